// GAT_39238821216304
// MI455X (gfx1250) — compile-verified
//
#include <hip/hip_runtime.h>
#include <math.h>

typedef float v2f __attribute__((ext_vector_type(2)));
typedef float v8f __attribute__((ext_vector_type(8)));

#define HID 128
#define OUTF 64
#define HEADS 2
#define NG 64
#define NEG_SLOPE 0.2f

// ---------------- init helpers ----------------
__global__ void init_val_kernel(float* __restrict__ p, int n, float v) {
    int i = blockIdx.x * blockDim.x + threadIdx.x;
    if (i < n) p[i] = v;
}

// ---------------- fp32 WMMA GEMM: C[M,128] = A[M,128] @ B[128,128] ----------------
// One block = 256 threads = 8 waves. Block computes a 16x128 output tile.
// A tile (16x128, 8KB) staged in LDS, shared by all 8 waves; each wave owns a
// 16-wide N slice and runs 32 x V_WMMA_F32_16X16X4_F32 over K=128.
// In-place C==A is safe: each block fully reads its own 16 A-rows into LDS
// (then __syncthreads) before writing C, and blocks touch disjoint rows.
__global__ __launch_bounds__(256) void gemm128_f32_wmma(
        const float* __restrict__ A, const float* __restrict__ B,
        float* __restrict__ C, int M) {
    __shared__ float As[16 * 128];
    const int mbase = blockIdx.x * 16;
    for (int i = threadIdx.x; i < 16 * 128; i += 256) {
        int r = i >> 7, c = i & 127;
        As[i] = (mbase + r < M) ? A[(size_t)(mbase + r) * HID + c] : 0.0f;
    }
    __syncthreads();

    const int wave = threadIdx.x >> 5;       // 0..7 -> N tile
    const int lane = threadIdx.x & 31;
    const int nbase = wave * 16;
    const int mrow = lane & 15;              // A: M = lane&15
    const int kgrp = (lane >> 4) << 1;       // lanes 16..31 hold K+2 pair

    v8f acc = {};
    for (int k = 0; k < 128; k += 4) {
        v2f a, b;
        a.x = As[mrow * 128 + k + kgrp];
        a.y = As[mrow * 128 + k + kgrp + 1];
        const int col = nbase + (lane & 15);
        b.x = B[(size_t)(k + kgrp) * HID + col];
        b.y = B[(size_t)(k + kgrp + 1) * HID + col];
        acc = __builtin_amdgcn_wmma_f32_16x16x4_f32(
            false, a, false, b, (short)0, acc, false, false);
    }

    // C/D layout: VGPR r -> M = r (lanes 0-15) / r+8 (lanes 16-31), N = lane&15
    const int col = nbase + (lane & 15);
    const int rbase = mbase + ((lane >> 4) << 3);
#pragma unroll
    for (int r = 0; r < 8; ++r) {
        int row = rbase + r;
        if (row < M) C[(size_t)row * HID + col] = acc[r];
    }
}

// ---------------- attention scores: a_src/a_dst [N,2] ----------------
// One wave per node. col = lane*4 spans [head*64 + c] exactly matching the
// flattened att_src/att_dst [2,64] layout; reduce within 16-lane head groups.
__global__ __launch_bounds__(256) void att_score_kernel(
        const float* __restrict__ h, const float* __restrict__ att_src,
        const float* __restrict__ att_dst, float* __restrict__ asrc,
        float* __restrict__ adst, int N) {
    const int wave = threadIdx.x >> 5;
    const int lane = threadIdx.x & 31;
    const int n = blockIdx.x * 8 + wave;
    if (n >= N) return;
    const int col = lane * 4;
    const float4 hv = *(const float4*)(h + (size_t)n * HID + col);
    const float4 sv = *(const float4*)(att_src + col);
    const float4 dv = *(const float4*)(att_dst + col);
    float ps = hv.x * sv.x + hv.y * sv.y + hv.z * sv.z + hv.w * sv.w;
    float pd = hv.x * dv.x + hv.y * dv.y + hv.z * dv.z + hv.w * dv.w;
#pragma unroll
    for (int m = 1; m < 16; m <<= 1) {
        ps += __shfl_xor(ps, m, 32);
        pd += __shfl_xor(pd, m, 32);
    }
    if ((lane & 15) == 0) {
        int head = lane >> 4;
        asrc[(size_t)n * HEADS + head] = ps;
        adst[(size_t)n * HEADS + head] = pd;
    }
}

// ---------------- segment-sum edge_attr by src + degree count ----------------
__global__ __launch_bounds__(256) void edge_sum_kernel(
        const int* __restrict__ srcA, const float* __restrict__ edge_attr,
        float* __restrict__ esum, float* __restrict__ deg, int E) {
    const int e = blockIdx.x * 2 + (threadIdx.x >> 7);
    const int c = threadIdx.x & 127;
    if (e >= E) return;
    const int s = srcA[e];
    atomicAdd(&esum[(size_t)s * HID + c], edge_attr[(size_t)e * HID + c]);
    if (c == 0) atomicAdd(&deg[s], 1.0f);
}

// ---------------- edge pass 1: leaky-relu score + segment max ----------------
__global__ void edge_emax_kernel(
        const int* __restrict__ srcA, const int* __restrict__ dstA,
        const float* __restrict__ asrc, const float* __restrict__ adst,
        float* __restrict__ ews, float* __restrict__ emax, int E, int N) {
    const int t = blockIdx.x * blockDim.x + threadIdx.x;
    if (t >= E + N) return;
    int s, d;
    if (t < E) { s = srcA[t]; d = dstA[t]; } else { s = t - E; d = s; }
#pragma unroll
    for (int hh = 0; hh < HEADS; ++hh) {
        float e = asrc[(size_t)s * HEADS + hh] + adst[(size_t)d * HEADS + hh];
        e = (e >= 0.0f) ? e : NEG_SLOPE * e;
        ews[(size_t)t * HEADS + hh] = e;
        atomicMax(&emax[(size_t)d * HEADS + hh], e);
    }
}

// ---------------- edge pass 2: exp + segment sum ----------------
__global__ void edge_exp_kernel(
        const int* __restrict__ dstA, float* __restrict__ ews,
        const float* __restrict__ emax, float* __restrict__ denom, int E, int N) {
    const int t = blockIdx.x * blockDim.x + threadIdx.x;
    if (t >= E + N) return;
    const int d = (t < E) ? dstA[t] : (t - E);
#pragma unroll
    for (int hh = 0; hh < HEADS; ++hh) {
        float ex = __expf(ews[(size_t)t * HEADS + hh] - emax[(size_t)d * HEADS + hh]);
        ews[(size_t)t * HEADS + hh] = ex;
        atomicAdd(&denom[(size_t)d * HEADS + hh], ex);
    }
}

// ---------------- edge pass 3: alpha-weighted gather/scatter ----------------
// One wave per edge; lane handles 4 contiguous channels (head = lane>>4).
__global__ __launch_bounds__(256) void message_kernel(
        const int* __restrict__ srcA, const int* __restrict__ dstA,
        const float* __restrict__ h, const float* __restrict__ ews,
        const float* __restrict__ denom, float* __restrict__ nf, int E, int N) {
    const int wave = threadIdx.x >> 5;
    const int lane = threadIdx.x & 31;
    const int t = blockIdx.x * 8 + wave;
    if (t >= E + N) return;
    int s, d;
    if (t < E) { s = srcA[t]; d = dstA[t]; } else { s = t - E; d = s; }
    const int col = lane * 4;
    const int head = lane >> 4;
    const float alpha = ews[(size_t)t * HEADS + head] /
                        (denom[(size_t)d * HEADS + head] + 1e-16f);
    const float4 hv = *(const float4*)(h + (size_t)s * HID + col);
    float* out = nf + (size_t)d * HID + col;
    atomicAdd(out + 0, hv.x * alpha);
    atomicAdd(out + 1, hv.y * alpha);
    atomicAdd(out + 2, hv.z * alpha);
    atomicAdd(out + 3, hv.w * alpha);
}

// ---------------- combine + sorted-batch pooling ----------------
// batch is sorted, so accumulate locally and flush one atomic per graph change.
__global__ __launch_bounds__(256) void combine_pool_kernel(
        const float* __restrict__ nf, const float* __restrict__ eg,
        const float* __restrict__ deg, const int* __restrict__ batch,
        const float* __restrict__ gat_bias, const float* __restrict__ edge_b,
        float* __restrict__ pooled, int N) {
    const int col = threadIdx.x & 127;
    const int sub = threadIdx.x >> 7;  // 0 or 1
    const int nstart = blockIdx.x * 64;
    const float gb = gat_bias[col];
    const float eb = edge_b[col];
    float acc = 0.0f;
    int cur = -1;
    for (int n = nstart + sub; n < nstart + 64 && n < N; n += 2) {
        int g = batch[n];
        if (g != cur) {
            if (cur >= 0) atomicAdd(&pooled[(size_t)cur * HID + col], acc);
            acc = 0.0f;
            cur = g;
        }
        acc += nf[(size_t)n * HID + col] + gb +
               eg[(size_t)n * HID + col] + deg[n] * eb;
    }
    if (cur >= 0) atomicAdd(&pooled[(size_t)cur * HID + col], acc);
}

// ---------------- tiny MLP: [64,128]@[128,32]+b1 -> @[32,1]+b2 ----------------
__global__ __launch_bounds__(128) void mlp_kernel(
        const float* __restrict__ pooled, const float* __restrict__ w1,
        const float* __restrict__ b1, const float* __restrict__ w2,
        const float* __restrict__ b2, float* __restrict__ out) {
    __shared__ float p[HID];
    __shared__ float hid[32];
    const int g = blockIdx.x;
    p[threadIdx.x] = pooled[(size_t)g * HID + threadIdx.x];
    __syncthreads();
    if (threadIdx.x < 32) {
        float a = b1[threadIdx.x];
        for (int c = 0; c < HID; ++c) a += p[c] * w1[c * 32 + threadIdx.x];
        hid[threadIdx.x] = a;
    }
    __syncthreads();
    if (threadIdx.x == 0) {
        float a = b2[0];
        for (int j = 0; j < 32; ++j) a += hid[j] * w2[j];
        out[g] = a;
    }
}

extern "C" void kernel_launch(void* const* d_in, const int* in_sizes, int n_in,
                              void* d_out, int out_size, void* d_ws, size_t ws_size,
                              hipStream_t stream) {
    const float* x        = (const float*)d_in[0];
    const int*   ei       = (const int*)d_in[1];
    const float* eattr    = (const float*)d_in[2];
    const int*   batch    = (const int*)d_in[3];
    const float* lin_w    = (const float*)d_in[4];
    const float* att_src  = (const float*)d_in[5];
    const float* att_dst  = (const float*)d_in[6];
    const float* gat_bias = (const float*)d_in[7];
    const float* edge_w   = (const float*)d_in[8];
    const float* edge_b   = (const float*)d_in[9];
    const float* w1       = (const float*)d_in[10];
    const float* b1       = (const float*)d_in[11];
    const float* w2       = (const float*)d_in[12];
    const float* b2       = (const float*)d_in[13];
    float* out = (float*)d_out;

    const int N = in_sizes[0] / HID;
    const int E = in_sizes[1] / 2;
    const int* srcA = ei;
    const int* dstA = ei + E;

    // workspace layout (floats)
    float* ws = (float*)d_ws;
    float* h      = ws; ws += (size_t)N * HID;       // x @ lin_w
    float* esum   = ws; ws += (size_t)N * HID;       // segsum(edge_attr) -> in-place GEMM
    float* nf     = ws; ws += (size_t)N * HID;       // GAT aggregation
    float* deg    = ws; ws += (size_t)N;             // out-degree (for edge_b)
    float* asrc   = ws; ws += (size_t)N * HEADS;
    float* adst   = ws; ws += (size_t)N * HEADS;
    float* emax   = ws; ws += (size_t)N * HEADS;
    float* denom  = ws; ws += (size_t)N * HEADS;
    float* ews    = ws; ws += (size_t)(E + N) * HEADS;  // per-edge score / exp
    float* pooled = ws; ws += (size_t)NG * HID;

    const int tot = E + N;

    // 1) inits: esum, nf, deg are contiguous -> single zero pass
    {
        int nz = 2 * N * HID + N;
        init_val_kernel<<<(nz + 255) / 256, 256, 0, stream>>>(esum, nz, 0.0f);
        init_val_kernel<<<(2 * N + 255) / 256, 256, 0, stream>>>(denom, 2 * N, 0.0f);
        init_val_kernel<<<(NG * HID + 255) / 256, 256, 0, stream>>>(pooled, NG * HID, 0.0f);
        init_val_kernel<<<(2 * N + 255) / 256, 256, 0, stream>>>(emax, 2 * N, -3.0e38f);
    }

    // 2) h = x @ lin_w   (fp32 WMMA)
    gemm128_f32_wmma<<<(N + 15) / 16, 256, 0, stream>>>(x, lin_w, h, N);

    // 3) attention scores
    att_score_kernel<<<(N + 7) / 8, 256, 0, stream>>>(h, att_src, att_dst, asrc, adst, N);

    // 4) segment-sum edge features by source node (+degree)
    edge_sum_kernel<<<(E + 1) / 2, 256, 0, stream>>>(srcA, eattr, esum, deg, E);

    // 5) esum = esum @ edge_w   (in-place, fp32 WMMA) == segsum(edge_attr @ edge_w); deg*edge_b added in step 9
    gemm128_f32_wmma<<<(N + 15) / 16, 256, 0, stream>>>(esum, edge_w, esum, N);

    // 6-8) edge softmax + message passing (self-loops appended as t >= E)
    edge_emax_kernel<<<(tot + 255) / 256, 256, 0, stream>>>(srcA, dstA, asrc, adst, ews, emax, E, N);
    edge_exp_kernel<<<(tot + 255) / 256, 256, 0, stream>>>(dstA, ews, emax, denom, E, N);
    message_kernel<<<(tot + 7) / 8, 256, 0, stream>>>(srcA, dstA, h, ews, denom, nf, E, N);

    // 9) combine node_feat + bias + edge aggregate + deg*edge_b, pool by sorted batch
    combine_pool_kernel<<<(N + 63) / 64, 256, 0, stream>>>(
        nf, esum, deg, batch, gat_bias, edge_b, pooled, N);

    // 10) MLP head
    mlp_kernel<<<NG, 128, 0, stream>>>(pooled, w1, b1, w2, b2, out);
}